// ReconGraph_40389872451946
// MI455X (gfx1250) — compile-verified
//
#include <hip/hip_runtime.h>
#include <stdint.h>

#define MM  128
#define NNW 128
#define PIX (MM * NNW)   // 16384 pixels; adj is PIX x PIX = 268,435,456 floats (1 GiB)

// ---------------------------------------------------------------------------
// Kernel 1: bulk zero-fill of the 1 GiB adjacency matrix.
// Store-bandwidth bound: 1 GiB / 23.3 TB/s ~= 46 us floor.
// Uses gfx1250 async LDS->global stores (ASYNCcnt) with non-temporal hint so
// we do not thrash the 192 MB L2 with a 1 GiB streaming write.
// Each workgroup zeroes 4 KB of LDS once, then every lane repeatedly issues
// global_store_async_from_lds_b128 from its private 16 B LDS slot to
// contiguous global addresses (each wave32 = 512 B contiguous burst).
// ---------------------------------------------------------------------------
__global__ void __launch_bounds__(256) fill_zero_async_kernel(float* __restrict__ out,
                                                              long long n4) {
  extern __shared__ unsigned char smem_raw[];
  float4* s = (float4*)smem_raw;
  s[threadIdx.x] = make_float4(0.0f, 0.0f, 0.0f, 0.0f);
  __syncthreads();

  // LDS aperture: flat address low 32 bits == LDS byte offset.
  unsigned lds_addr = (unsigned)(unsigned long long)(uintptr_t)&s[threadIdx.x];

  float4* out4 = (float4*)out;
  long long stride = (long long)gridDim.x * (long long)blockDim.x;
  for (long long idx = (long long)blockIdx.x * blockDim.x + threadIdx.x;
       idx < n4; idx += stride) {
    unsigned long long gaddr = (unsigned long long)(out4 + idx);
    asm volatile("global_store_async_from_lds_b128 %0, %1, off th:TH_STORE_NT"
                 :
                 : "v"(gaddr), "v"(lds_addr)
                 : "memory");
  }
  // Drain ASYNCcnt before the workgroup's LDS can be retired.
  asm volatile("s_wait_asynccnt 0x0" ::: "memory");
}

// ---------------------------------------------------------------------------
// Kernel 2: diagonal-neighbor similarity scatter.
// One thread per pixel (16384 threads). Reads center + up to 4 diagonal
// neighbors, writes 1.0f where |diff| <= threshold. Each (src,dst) pair is
// unique across the 4 shifts, and the canvas is already zero, so writing only
// the true entries is exactly adj.at[src,dst].set(mask).
// ---------------------------------------------------------------------------
__global__ void __launch_bounds__(256) edge_kernel(const float* __restrict__ d,
                                                   const float* __restrict__ thr_p,
                                                   float* __restrict__ out) {
  int idx = blockIdx.x * blockDim.x + threadIdx.x;
  if (idx >= PIX) return;

  int i = idx >> 7;       // row (N == 128)
  int j = idx & 127;      // col
  float thr = thr_p[0];
  float c = d[idx];
  size_t rowbase = (size_t)idx * (size_t)PIX;

  const int dys[4] = {-1, -1, 1, 1};
  const int dxs[4] = {-1, 1, -1, 1};
#pragma unroll
  for (int k = 0; k < 4; ++k) {
    int ii = i + dys[k];
    int jj = j + dxs[k];
    if ((unsigned)ii < (unsigned)MM && (unsigned)jj < (unsigned)NNW) {
      int dst = ii * NNW + jj;
      float v = d[dst];
      if (fabsf(v - c) <= thr) {
        out[rowbase + (size_t)dst] = 1.0f;
      }
    }
  }
}

// ---------------------------------------------------------------------------
// Launcher: zero-fill first, then scatter (stream order serializes them).
// ---------------------------------------------------------------------------
extern "C" void kernel_launch(void* const* d_in, const int* in_sizes, int n_in,
                              void* d_out, int out_size, void* d_ws, size_t ws_size,
                              hipStream_t stream) {
  const float* d_noised = (const float*)d_in[0];   // [1,128,128] float32
  const float* thr      = (const float*)d_in[1];   // scalar float32
  float* out            = (float*)d_out;           // 16384*16384 floats

  long long n  = (long long)out_size;              // 268,435,456
  long long n4 = n >> 2;                           // 67,108,864 float4 stores

  // 4096 blocks * 256 threads * 64 iters == n4 exactly (guarded anyway).
  dim3 fill_grid(4096), fill_block(256);
  hipLaunchKernelGGL(fill_zero_async_kernel, fill_grid, fill_block,
                     256 * sizeof(float4), stream, out, n4);

  dim3 edge_grid((PIX + 255) / 256), edge_block(256);
  hipLaunchKernelGGL(edge_kernel, edge_grid, edge_block, 0, stream,
                     d_noised, thr, out);
}